// GraphwiseKLLoss_11398843204110
// MI455X (gfx1250) — compile-verified
//
#include <hip/hip_runtime.h>

#define EPS 1e-8f
#define LN2 0.69314718055994530942f

typedef __attribute__((ext_vector_type(2))) float v2f;
typedef __attribute__((ext_vector_type(8))) float v8f;

// Exchange with lane^16 in one LDS-pipe op: ds_swizzle group-of-32,
// SWAPX16 pattern (xor=0x10, or=0x00, and=0x1f) => offset 0x401F.
__device__ __forceinline__ float swap_half16(float x) {
    return __int_as_float(__builtin_amdgcn_ds_swizzle(__float_as_int(x), 0x401f));
}

// Wave32 all-lane sum on the matrix pipe:
// A(16x4): lane L supplies A[L%16][2*(L/16)] = v (other K slots 0).
// B(4x16): all ones.  D[m][n] = v[m] + v[m+16] for every n.
// Summing the 8 D VGPRs at lane l gives 8 consecutive row-sums; adding the
// value from lane l^16 completes the 32-lane total (result in all lanes).
// Requires EXEC == all 1s at the call site.
__device__ __forceinline__ float wave_sum32(float v) {
    v2f a; a.x = v;    a.y = 0.0f;
    v2f b; b.x = 1.0f; b.y = 1.0f;
    v8f c = {};
    v8f d = __builtin_amdgcn_wmma_f32_16x16x4_f32(
        /*neg_a=*/false, a, /*neg_b=*/false, b,
        /*c_mod=*/(short)0, c, /*reuse_a=*/false, /*reuse_b=*/false);
    float s = d[0] + d[1] + d[2] + d[3] + d[4] + d[5] + d[6] + d[7];
    s += swap_half16(s);
    return s;
}

// Fused single pass: one wave per segment g accumulates
//   S1  = sum p,            p = max(y_true,0)*max(weight,EPS)
//   S23 = sum p*ln(p/q),    q = max(y_pred,EPS)
// then  segkl[g] = (S23 - S1*ln(Pm)) / Pm,  Pm = max(S1, EPS)
// which equals sum_i pn*(ln(max(pn,EPS)) - ln q), pn = p/Pm, up to the EPS
// clamp on pn inside the log (affects only p < EPS*Pm, contribution O(1e-11)).
__global__ void __launch_bounds__(256)
gkl_segment_kernel(const float* __restrict__ y_pred,
                   const float* __restrict__ y_true,
                   const float* __restrict__ weight,
                   const int*   __restrict__ ptr,
                   float*       __restrict__ segkl,
                   int numSeg) {
    int wave = (blockIdx.x * blockDim.x + threadIdx.x) >> 5;
    int lane = threadIdx.x & 31;
    if (wave >= numSeg) return;              // wave-uniform exit, EXEC stays full
    int beg = ptr[wave];
    int end = ptr[wave + 1];

    float s1  = 0.0f;
    float s23 = 0.0f;
    for (int i = beg + lane; i < end; i += 32) {
        __builtin_prefetch(&y_pred[i + 256], 0, 0);  // global_prefetch_b8 (speculative)
        __builtin_prefetch(&y_true[i + 256], 0, 0);
        __builtin_prefetch(&weight[i + 256], 0, 0);
        float p = fmaxf(y_true[i], 0.0f) * fmaxf(weight[i], EPS);
        float q = fmaxf(y_pred[i], EPS);
        s1  += p;
        // p * (ln p - ln q); clamp keeps p==0 contributing exactly 0.
        s23 += p * (LN2 * (__log2f(fmaxf(p, 1e-30f)) - __log2f(q)));
    }

    // Two back-to-back WMMA wave reductions (compiler inserts hazard NOPs).
    float S1  = wave_sum32(s1);
    float S23 = wave_sum32(s23);

    if (lane == 0) {
        float Pm = fmaxf(S1, EPS);
        float lnPm = LN2 * __log2f(Pm);
        segkl[wave] = (S23 - S1 * lnPm) / Pm;
    }
}

// Final deterministic tree reduce of segkl -> mean over graphs.
__global__ void __launch_bounds__(1024)
gkl_finalize_kernel(const float* __restrict__ segkl,
                    int numSeg,
                    float* __restrict__ out) {
    __shared__ float smem[32];
    int tid  = threadIdx.x;
    int lane = tid & 31;
    int wid  = tid >> 5;
    float acc = 0.0f;
    for (int i = tid; i < numSeg; i += 1024) {
        __builtin_prefetch(&segkl[i + 4096], 0, 0);
        acc += segkl[i];
    }
    float s = wave_sum32(acc);               // EXEC all-1s: every thread alive
    if (lane == 0) smem[wid] = s;
    __syncthreads();
    if (wid == 0) {
        float v = smem[lane];                // 32 waves -> 32 valid partials
        float t = wave_sum32(v);
        if (lane == 0) out[0] = t / (float)(numSeg > 0 ? numSeg : 1);
    }
}

extern "C" void kernel_launch(void* const* d_in, const int* in_sizes, int n_in,
                              void* d_out, int out_size, void* d_ws, size_t ws_size,
                              hipStream_t stream) {
    const float* y_pred = (const float*)d_in[0];
    const float* y_true = (const float*)d_in[1];
    const float* weight = (const float*)d_in[2];
    const int*   ptr    = (const int*)d_in[3];
    int numSeg = in_sizes[3] - 1;            // 65536

    float* segkl = (float*)d_ws;             // [numSeg] = 256 KB scratch

    int wavesPerBlock = 8;                   // 256 threads = 8 wave32
    int blocks = (numSeg + wavesPerBlock - 1) / wavesPerBlock;

    gkl_segment_kernel<<<blocks, 256, 0, stream>>>(y_pred, y_true, weight, ptr,
                                                   segkl, numSeg);
    gkl_finalize_kernel<<<1, 1024, 0, stream>>>(segkl, numSeg, (float*)d_out);
}